// CatmullRomSplines_85993835201257
// MI455X (gfx1250) — compile-verified
//
#include <hip/hip_runtime.h>
#include <cmath>

typedef float v2f __attribute__((ext_vector_type(2)));
typedef float v8f __attribute__((ext_vector_type(8)));
typedef int   v2i __attribute__((ext_vector_type(2)));

// Per-lane A-matrix operand for V_WMMA_F32_16X16X4_F32.
// 32-bit 16x4 A layout (ISA 7.12.2): lanes 0-15 hold K=0 (VGPR0), K=1 (VGPR1)
// for row M=lane; lanes 16-31 hold K=2,3 for row M=lane-16.
// Rows 0..3 = Catmull-Rom basis matrix A, rows 4..15 = 0 (padding).
__constant__ float A_LANE_TAB[64] = {
    // lanes 0..3 : A[m][0], A[m][1]
    -0.5f,  1.5f,
     1.0f, -2.5f,
    -0.5f,  0.0f,
     0.0f,  1.0f,
    // lanes 4..15 : zero-padded rows
    0,0, 0,0, 0,0, 0,0, 0,0, 0,0, 0,0, 0,0, 0,0, 0,0, 0,0, 0,0,
    // lanes 16..19 : A[m][2], A[m][3]
    -1.5f,  0.5f,
     2.0f, -0.5f,
     0.5f,  0.0f,
     0.0f,  0.0f,
    // lanes 20..31 : zero-padded rows
    0,0, 0,0, 0,0, 0,0, 0,0, 0,0, 0,0, 0,0, 0,0, 0,0, 0,0, 0,0
};

// Each wave processes 32 points. For group g (16 points), the WMMA computes
// D = Apad(16x4) @ B(4x16) where column n of B holds the 4 control-point
// values (one channel) of point base+g*16+n:
//   B[0][n]=CP_locs[i-1,j]  B[1][n]=CP_locs[i,j]
//   B[2][n]=CP_locs[i,j+1]  B[3][n]=CP_locs[i-1,j+1]
// Lane n supplies (B0,B1); lane n+16 supplies (B2,B3).  Result: lane n
// (n<16) receives x1[0..3] for its column in d[0..3] -> lane-local Horner.
__global__ __launch_bounds__(256) void catmull_wmma_kernel(
    const float* __restrict__ x_input,
    const float* __restrict__ CP_locs,
    const int*   __restrict__ CP_idx,
    float*       __restrict__ out,
    int N, int G)
{
    const int lane = threadIdx.x & 31;
    const int wave = (blockIdx.x * blockDim.x + threadIdx.x) >> 5;
    const int base = wave * 32;
    const int half = lane >> 4;   // 0: lanes 0-15, 1: lanes 16-31
    const int col  = lane & 15;

    // A operand (2 VGPRs per lane), constant per kernel.
    v2f a;
    a.x = A_LANE_TAB[2 * lane + 0];
    a.y = A_LANE_TAB[2 * lane + 1];

    v2f b0[2], b1[2], t[2];
    int pst[2];

#pragma unroll
    for (int g = 0; g < 2; ++g) {
        int p  = base + g * 16 + col;
        pst[g] = p;
        int pc = p < N ? p : N - 1;   // clamp so EXEC stays all-1s for WMMA

        // Streamed inputs: non-temporal (no reuse; keep L2 for CP_locs).
        v2i idx = __builtin_nontemporal_load((const v2i*)(CP_idx  + 2 * (size_t)pc));
        v2f xin = __builtin_nontemporal_load((const v2f*)(x_input + 2 * (size_t)pc));
        int i = idx.x, j = idx.y;

        // Cooperative gather of the 4 control points across the lane pair.
        int bcol = j + half;
        int rA   = half ? i     : i - 1;   // half0: CP0 row; half1: CP2 row
        int rB   = half ? i - 1 : i;       // half0: CP1 row; half1: CP3 row
        v2f e0 = *(const v2f*)(CP_locs + 2 * ((size_t)rA * G + bcol));
        v2f e1 = *(const v2f*)(CP_locs + 2 * ((size_t)rB * G + bcol));

        v2f bb0 = {e0.x, e1.x};   // channel 0 B operand
        v2f bb1 = {e0.y, e1.y};   // channel 1 B operand
        b0[g] = bb0;
        b1[g] = bb1;

        // corner = CP_locs[i][j] == e1 in lanes 0-15 (only those lanes use t)
        t[g] = xin - e1;
    }

    v8f c = {};
    // x1 = A @ CP_ch2 for 16 columns per WMMA (4 WMMAs = 32 pts x 2 ch)
    v8f d00 = __builtin_amdgcn_wmma_f32_16x16x4_f32(false, a, false, b0[0], (short)0, c, false, false);
    v8f d01 = __builtin_amdgcn_wmma_f32_16x16x4_f32(false, a, false, b1[0], (short)0, c, false, false);
    v8f d10 = __builtin_amdgcn_wmma_f32_16x16x4_f32(false, a, false, b0[1], (short)0, c, false, false);
    v8f d11 = __builtin_amdgcn_wmma_f32_16x16x4_f32(false, a, false, b1[1], (short)0, c, false, false);

    // Lane n (n<16) owns points base+n and base+16+n: Horner + store.
    if (half == 0) {
#pragma unroll
        for (int g = 0; g < 2; ++g) {
            v8f dc0 = g ? d10 : d00;
            v8f dc1 = g ? d11 : d01;
            float t0 = t[g].x, t1 = t[g].y;
            float o0 = fmaf(fmaf(fmaf(dc0[0], t0, dc0[1]), t0, dc0[2]), t0, dc0[3]);
            float o1 = fmaf(fmaf(fmaf(dc1[0], t1, dc1[1]), t1, dc1[2]), t1, dc1[3]);
            int p = pst[g];
            if (p < N) {
                v2f o = {o0, o1};
                __builtin_nontemporal_store(o, (v2f*)(out + 2 * (size_t)p));
            }
        }
    }
}

extern "C" void kernel_launch(void* const* d_in, const int* in_sizes, int n_in,
                              void* d_out, int out_size, void* d_ws, size_t ws_size,
                              hipStream_t stream) {
    const float* x_input = (const float*)d_in[0];
    const float* CP_locs = (const float*)d_in[1];
    const int*   CP_idx  = (const int*)d_in[2];
    float* out = (float*)d_out;

    int N = in_sizes[0] / 2;                                  // (N,2) f32
    int G = (int)(std::sqrt((double)(in_sizes[1] / 2)) + 0.5); // (G,G,2) f32

    int blocks = (N + 255) / 256;   // 256 thr = 8 waves = 256 points/block
    catmull_wmma_kernel<<<blocks, 256, 0, stream>>>(x_input, CP_locs, CP_idx, out, N, G);
}